// PSDLoss2_83081847373855
// MI455X (gfx1250) — compile-verified
//
#include <hip/hip_runtime.h>
#include <math.h>
#include <stdint.h>

// PSD loss on MI455X (gfx1250): DFT-as-matmul on the WMMA matrix pipe.
// F = W * G * W^T with W = C - iS (512x512 symmetric cos/sin tables).
// Row pass:    Tre = G*C, Tim = -(G*S)          (2 f32 WMMA GEMMs, shared A)
// Column pass: Fre = C*Tre + S*Tim, Fim = C*Tim - S*Tre  (4 f32 WMMA accums)
//   GEMM2 stages its shared B panels (Tre/Tim) into LDS with
//   global_load_async_to_lds_b64 (ASYNCcnt path), 4 waves share each panel.
// Fused epilogue: log-magnitude + fftshift + radial binning via LDS atomics
// (never materializes the 268 MB complex spectrum).
// Workspace use: ~53 MB (chunked 16 images at a time, 8 chunks of 128 total).

#define WD 512
#define NIMG 128            // 64 pred + 64 target grayscale images
#define CH 16               // images per chunk
#define NCHUNK (NIMG / CH)
#define NBINS 716           // int(sqrt(2)*506) + 1, center hardcoded at (5,5)
#define EPSF 1e-8f
#define KC 16               // K-slab staged in LDS per iteration (4 WMMA steps)

typedef __attribute__((ext_vector_type(2))) float v2f;
typedef __attribute__((ext_vector_type(8))) float v8f;

static __device__ __forceinline__ v8f wmma4(v2f a, v2f b, v8f c) {
  // V_WMMA_F32_16X16X4_F32 : D(16x16 f32) = A(16x4) * B(4x16) + C
  return __builtin_amdgcn_wmma_f32_16x16x4_f32(
      /*neg_a=*/false, a, /*neg_b=*/false, b,
      /*c_mod=*/(short)0, c, /*reuse_a=*/false, /*reuse_b=*/false);
}

// LDS flat pointer -> 32-bit LDS byte offset (aperture bits live in [63:32]).
static __device__ __forceinline__ uint32_t lds_off(const void* p) {
  return (uint32_t)(uintptr_t)p;
}

// Async global -> LDS copy of 8 bytes per lane (GV mode, tracked by ASYNCcnt).
static __device__ __forceinline__ void async_copy_b64(uint32_t dst_lds_off,
                                                      const float* src) {
  uint64_t ga = (uint64_t)(uintptr_t)src;
  asm volatile("global_load_async_to_lds_b64 %0, %1, off"
               :: "v"(dst_lds_off), "v"(ga)
               : "memory");
}

static __device__ __forceinline__ void wait_async0() {
  asm volatile("s_wait_asynccnt 0x0" ::: "memory");
}

// ---------------------------------------------------------------- utilities

__global__ __launch_bounds__(256) void zero_kernel(float* __restrict__ p, int n) {
  int i = blockIdx.x * 256 + threadIdx.x;
  if (i < n) p[i] = 0.0f;
}

// Ct[k][n] = cos(2*pi*k*n/512), St[k][n] = sin(...); exact via (k*n mod 512).
__global__ __launch_bounds__(256) void table_kernel(float* __restrict__ Ct,
                                                    float* __restrict__ St) {
  int idx = blockIdx.x * 256 + threadIdx.x;
  if (idx >= WD * WD) return;
  int k = idx >> 9, n = idx & (WD - 1);
  int m = (k * n) & (WD - 1);
  float ang = (float)m * (6.28318530717958647692f / (float)WD);
  Ct[idx] = cosf(ang);
  St[idx] = sinf(ang);
}

// Radial bin population counts (shift is a permutation -> same histogram).
__global__ __launch_bounds__(256) void count_kernel(float* __restrict__ counts) {
  int idx = blockIdx.x * 256 + threadIdx.x;
  if (idx >= WD * WD) return;
  int y = idx >> 9, x = idx & (WD - 1);
  float dx = (float)x - 5.0f, dy = (float)y - 5.0f;
  int b = (int)sqrtf(dx * dx + dy * dy);
  atomicAdd(&counts[b], 1.0f);
}

// gray = 0.2989 R + 0.587 G + 0.114 B for one chunk of CH images.
__global__ __launch_bounds__(256) void gray_kernel(const float* __restrict__ pred,
                                                   const float* __restrict__ targ,
                                                   float* __restrict__ gray,
                                                   int chunk) {
  int idx = blockIdx.x * 256 + threadIdx.x;
  if (idx >= CH * WD * WD) return;
  int j = idx / (WD * WD);
  int pix = idx - j * (WD * WD);
  int Bg = chunk * CH + j;
  const float* src = (Bg < 64) ? (pred + (size_t)Bg * 3 * WD * WD)
                               : (targ + (size_t)(Bg - 64) * 3 * WD * WD);
  float r = src[pix];
  float g = src[(size_t)WD * WD + pix];
  float b = src[2 * (size_t)WD * WD + pix];
  gray[idx] = 0.2989f * r + 0.587f * g + 0.114f * b;
}

// ------------------------------------------------ GEMM1: row DFT (real input)
// M = CH*512 (batch rows), N = 512 (k_x), K = 512 (n).
// Tre = G*C ; Tim = -(G*S). One wave per 16x16 tile; A fragment shared by the
// two WMMA accumulators.
__global__ __launch_bounds__(32) void gemm1_kernel(const float* __restrict__ gray,
                                                   const float* __restrict__ Ct,
                                                   const float* __restrict__ St,
                                                   float* __restrict__ Tre,
                                                   float* __restrict__ Tim) {
  const int lane = threadIdx.x;
  const int half = lane >> 4;      // 0/1 : K-pair select (ISA 16x4 A layout)
  const int lr   = lane & 15;      // row (A) / col (B,C,D) within tile
  const int tileN = blockIdx.x * 16;
  const int tileM = blockIdx.y * 16;
  const float* Arow = gray + (size_t)(tileM + lr) * WD;

  v8f accC = {};
  v8f accS = {};
  for (int k0 = 0; k0 < WD; k0 += 4) {
    int ka = k0 + 2 * half;
    v2f a;  a.x = Arow[ka];                              a.y = Arow[ka + 1];
    v2f bc; bc.x = Ct[(size_t)ka * WD + tileN + lr];     bc.y = Ct[(size_t)(ka + 1) * WD + tileN + lr];
    v2f bs; bs.x = St[(size_t)ka * WD + tileN + lr];     bs.y = St[(size_t)(ka + 1) * WD + tileN + lr];
    accC = wmma4(a, bc, accC);
    accS = wmma4(a, bs, accS);
  }
#pragma unroll
  for (int v = 0; v < 8; ++v) {
    size_t row = (size_t)tileM + v + 8 * half;     // D[v+8h][lr]
    size_t idx = row * WD + tileN + lr;
    Tre[idx] =  accC[v];
    Tim[idx] = -accS[v];
  }
}

// ---------------------------------------- GEMM2: column DFT + fused binning
// Per image: M = k_y, N = k_x, K = y (all 512).
// 4 waves per workgroup cover a 64x16 macro-tile; the shared B panels
// (Tre/Tim, KCx16) are staged in LDS via async global->LDS copies.
// Fre = C*Tre + S*Tim ; Fim = C*Tim - S*Tre.
// Epilogue: mag = log(sqrt((Fre+e)^2+(Fim+e)^2)+e), fftshift, radial bin into
// a shared LDS histogram, sparse-flush to global per-image bin sums.
__global__ __launch_bounds__(128) void gemm2_kernel(const float* __restrict__ Ct,
                                                    const float* __restrict__ St,
                                                    const float* __restrict__ Tre,
                                                    const float* __restrict__ Tim,
                                                    float* __restrict__ binsum,
                                                    int chunk) {
  __shared__ float bins[NBINS];
  __shared__ float bre[KC][16];
  __shared__ float bim[KC][16];

  const int tid  = threadIdx.x;        // 0..127
  const int wave = tid >> 5;           // 0..3 : M sub-tile
  const int lane = tid & 31;
  const int half = lane >> 4;
  const int lr   = lane & 15;
  const int tileN = blockIdx.x * 16;          // k_x
  const int tileM = blockIdx.y * 64;          // k_y macro-tile
  const int rowM  = tileM + wave * 16;        // this wave's 16 rows
  const int j  = blockIdx.z;                  // image within chunk
  const int Bg = chunk * CH + j;

  for (int i = tid; i < NBINS; i += 128) bins[i] = 0.0f;

  const float* Bre = Tre + (size_t)j * WD * WD;
  const float* Bim = Tim + (size_t)j * WD * WD;
  const float* ArowC = Ct + (size_t)(rowM + lr) * WD;
  const float* ArowS = St + (size_t)(rowM + lr) * WD;

  // Each of the 128 threads stages one b64 (2 floats) per tensor per K-slab.
  const int srow = tid >> 3;             // 0..15 : k-row within slab
  const int scol = (tid & 7) * 2;        // 0,2,..,14 : column pair
  const uint32_t dre = lds_off(&bre[srow][scol]);
  const uint32_t dim = lds_off(&bim[srow][scol]);

  v8f aRR = {}, aSI = {}, aCI = {}, aSR = {};
  for (int k0 = 0; k0 < WD; k0 += KC) {
    __syncthreads();   // previous slab fully consumed before overwrite
    async_copy_b64(dre, Bre + (size_t)(k0 + srow) * WD + tileN + scol);
    async_copy_b64(dim, Bim + (size_t)(k0 + srow) * WD + tileN + scol);
    wait_async0();     // this wave's async copies done (ASYNCcnt==0)
    __syncthreads();   // all waves' copies visible

#pragma unroll
    for (int kk = 0; kk < KC / 4; ++kk) {
      int ka = k0 + kk * 4 + 2 * half;     // global k for A
      int kl = kk * 4 + 2 * half;          // local k for LDS B
      v2f ac; ac.x = ArowC[ka];       ac.y = ArowC[ka + 1];
      v2f as; as.x = ArowS[ka];       as.y = ArowS[ka + 1];
      v2f br; br.x = bre[kl][lr];     br.y = bre[kl + 1][lr];
      v2f bi; bi.x = bim[kl][lr];     bi.y = bim[kl + 1][lr];
      aRR = wmma4(ac, br, aRR);
      aSI = wmma4(as, bi, aSI);
      aCI = wmma4(ac, bi, aCI);
      aSR = wmma4(as, br, aSR);
    }
  }

#pragma unroll
  for (int v = 0; v < 8; ++v) {
    float fre = aRR[v] + aSI[v] + EPSF;
    float fim = aCI[v] - aSR[v] + EPSF;
    float mag = logf(sqrtf(fre * fre + fim * fim) + EPSF);
    int ky = rowM + v + 8 * half;
    int kx = tileN + lr;
    int ys = (ky + WD / 2) & (WD - 1);    // fftshift (even W)
    int xs = (kx + WD / 2) & (WD - 1);
    float dx = (float)xs - 5.0f, dy = (float)ys - 5.0f;
    int bin = (int)sqrtf(dx * dx + dy * dy);
    atomicAdd(&bins[bin], mag);           // ds_add_f32
  }
  __syncthreads();
  for (int i = tid; i < NBINS; i += 128) {
    float s = bins[i];
    if (s != 0.0f) atomicAdd(&binsum[(size_t)Bg * NBINS + i], s);
  }
}

// ------------------------------------- per-image profile + min-max normalize
__global__ __launch_bounds__(256) void profile_kernel(float* __restrict__ binsum,
                                                      const float* __restrict__ counts) {
  const int Bg = blockIdx.x;
  const int tid = threadIdx.x;
  float* p = binsum + (size_t)Bg * NBINS;
  __shared__ float smn[256], smx[256];
  float vals[3];
  float mn = 3.4e38f, mx = -3.4e38f;
#pragma unroll
  for (int t = 0; t < 3; ++t) {
    int i = tid + t * 256;
    float v = 0.0f;
    if (i < NBINS) {
      v = p[i] / (counts[i] + 1e-10f);
      mn = fminf(mn, v);
      mx = fmaxf(mx, v);
    }
    vals[t] = v;
  }
  smn[tid] = mn; smx[tid] = mx;
  __syncthreads();
  for (int s = 128; s > 0; s >>= 1) {
    if (tid < s) {
      smn[tid] = fminf(smn[tid], smn[tid + s]);
      smx[tid] = fmaxf(smx[tid], smx[tid + s]);
    }
    __syncthreads();
  }
  mn = smn[0]; mx = smx[0];
  float inv = 1.0f / (mx - mn);
#pragma unroll
  for (int t = 0; t < 3; ++t) {
    int i = tid + t * 256;
    if (i < NBINS) p[i] = (vals[t] - mn) * inv;
  }
}

// ------------------------------------------------------- final MSE reduction
__global__ __launch_bounds__(256) void mse_kernel(const float* __restrict__ binsum,
                                                  float* __restrict__ out) {
  const int tid = threadIdx.x;
  const float* P = binsum;                          // images 0..63 (pred)
  const float* T = binsum + (size_t)64 * NBINS;     // images 64..127 (target)
  float s = 0.0f;
  for (int i = tid; i < 64 * NBINS; i += 256) {
    float d = P[i] - T[i];
    s += d * d;
  }
  __shared__ float sm[256];
  sm[tid] = s;
  __syncthreads();
  for (int w = 128; w > 0; w >>= 1) {
    if (tid < w) sm[tid] += sm[tid + w];
    __syncthreads();
  }
  if (tid == 0) out[0] = sm[0] / (64.0f * (float)NBINS);
}

extern "C" void kernel_launch(void* const* d_in, const int* in_sizes, int n_in,
                              void* d_out, int out_size, void* d_ws, size_t ws_size,
                              hipStream_t stream) {
  (void)in_sizes; (void)n_in; (void)out_size; (void)ws_size;
  const float* pred = (const float*)d_in[0];
  const float* targ = (const float*)d_in[1];
  float* out = (float*)d_out;

  // Workspace layout (floats); total ~52.8 MB.
  float* ws     = (float*)d_ws;
  float* Ct     = ws;                       // 512*512
  float* St     = Ct + WD * WD;             // 512*512
  float* counts = St + WD * WD;             // NBINS (padded to 768)
  float* binsum = counts + 768;             // NIMG*NBINS
  float* gray   = binsum + NIMG * NBINS;    // CH*512*512
  float* Tre    = gray + (size_t)CH * WD * WD;
  float* Tim    = Tre  + (size_t)CH * WD * WD;

  const int pixPerChunk = CH * WD * WD;

  // Setup: tables, zeroed accumulators, radial counts (re-done every launch
  // for determinism; harness does not re-poison between replays).
  table_kernel<<<(WD * WD + 255) / 256, 256, 0, stream>>>(Ct, St);
  zero_kernel<<<(NIMG * NBINS + 255) / 256, 256, 0, stream>>>(binsum, NIMG * NBINS);
  zero_kernel<<<(NBINS + 255) / 256, 256, 0, stream>>>(counts, NBINS);
  count_kernel<<<(WD * WD + 255) / 256, 256, 0, stream>>>(counts);

  for (int c = 0; c < NCHUNK; ++c) {
    gray_kernel<<<(pixPerChunk + 255) / 256, 256, 0, stream>>>(pred, targ, gray, c);
    gemm1_kernel<<<dim3(WD / 16, (CH * WD) / 16), 32, 0, stream>>>(gray, Ct, St, Tre, Tim);
    gemm2_kernel<<<dim3(WD / 16, WD / 64, CH), 128, 0, stream>>>(Ct, St, Tre, Tim, binsum, c);
  }

  profile_kernel<<<NIMG, 256, 0, stream>>>(binsum, counts);
  mse_kernel<<<1, 256, 0, stream>>>(binsum, out);
}